// GRANDConv_82772609728555
// MI455X (gfx1250) — compile-verified
//
#include <hip/hip_runtime.h>
#include <math.h>

#define NODES 50000
#define EDGES 600000
#define DIM   128
#define ORD   4

typedef float v2f __attribute__((ext_vector_type(2)));
typedef float v8f __attribute__((ext_vector_type(8)));

#define ATOMIC_ADD_F32(p, v) \
  __hip_atomic_fetch_add((p), (v), __ATOMIC_RELAXED, __HIP_MEMORY_SCOPE_AGENT)

// float atomic-max via monotone int/uint bit ordering (init must be -inf)
__device__ __forceinline__ void atomicMaxFloat(float* addr, float v) {
  if (v >= 0.0f)
    atomicMax((int*)addr, __float_as_int(v));
  else
    atomicMin((unsigned int*)addr, (unsigned int)__float_as_int(v));
}

__global__ void k_zero(float* __restrict__ p, int n) {
  int i = blockIdx.x * blockDim.x + threadIdx.x;
  if (i < n) p[i] = 0.0f;
}

__global__ void k_deg(const int* __restrict__ dst, float* __restrict__ deg, int n) {
  int i = blockIdx.x * blockDim.x + threadIdx.x;
  if (i < n) ATOMIC_ADD_F32(&deg[dst[i]], 1.0f);
}

__global__ void k_norm(float* __restrict__ nrm, int n) {
  int i = blockIdx.x * blockDim.x + threadIdx.x;
  if (i < n) nrm[i] = rsqrtf(fmaxf(nrm[i], 1.0f));
}

// x = feats ; y = feats  (float4 streams)
__global__ void k_init(const float* __restrict__ feats, float* __restrict__ x,
                       float* __restrict__ y, int n4) {
  int i = blockIdx.x * blockDim.x + threadIdx.x;
  if (i < n4) {
    float4 v = ((const float4*)feats)[i];
    ((float4*)x)[i] = v;
    ((float4*)y)[i] = v;
  }
}

// One wave per 16 nodes:
//  - zero h rows, init m = -inf, s = 0 for the tile
//  - a_src/a_dst = norm * (X @ [w_src | w_dst | 0...])   via V_WMMA_F32_16X16X4_F32
__global__ void __launch_bounds__(256)
k_node_dots(const float* __restrict__ x, const float* __restrict__ w,
            const float* __restrict__ nrm,
            float* __restrict__ a_s, float* __restrict__ a_d,
            float* __restrict__ h, float* __restrict__ m, float* __restrict__ s,
            int n_nodes) {
  int gtid = blockIdx.x * blockDim.x + threadIdx.x;
  int wave = gtid >> 5;
  int lane = gtid & 31;
  int base = wave * 16;
  if (base >= n_nodes) return;        // whole-wave uniform exit: EXEC stays all-1 for WMMA
  int nl = lane & 15;
  int hi = lane >> 4;

  // zero h tile (16 rows x 128 f32), per-lane float4 stores
  float4 z4 = make_float4(0.f, 0.f, 0.f, 0.f);
  float* hp = h + (size_t)base * DIM;
#pragma unroll
  for (int j = 0; j < 16; ++j)
    *(float4*)(hp + j * DIM + lane * 4) = z4;

  if (hi == 0) {
    m[base + nl] = __int_as_float(0xff800000);  // -inf
    s[base + nl] = 0.0f;
  }

  // A tile: row = base+nl, K-pair selected by hi (ISA 32-bit A 16x4 layout)
  const float* xrow = x + (size_t)(base + nl) * DIM;
  v8f c = {};
#pragma unroll
  for (int k0 = 0; k0 < DIM; k0 += 4) {
    int kk = k0 + 2 * hi;
    v2f a;
    a.x = xrow[kk];
    a.y = xrow[kk + 1];
    // B tile 4x16: col 0 = w_src slice, col 1 = w_dst slice, cols 2..15 = 0
    const float* wp = w + (nl & 1) * DIM + kk;  // valid address for every lane
    float bx = wp[0], by = wp[1];
    v2f b;
    b.x = (nl < 2) ? bx : 0.0f;
    b.y = (nl < 2) ? by : 0.0f;
    c = __builtin_amdgcn_wmma_f32_16x16x4_f32(false, a, false, b, (short)0, c,
                                              false, false);
  }

  // D layout: VGPR r, lanes 0-15 -> (M=r, N=lane); lanes 16-31 -> (M=8+r, N=lane-16)
  // column 0 (lanes 0,16) = x·w_src ; column 1 (lanes 1,17) = x·w_dst
  if (nl < 2) {
    float* outp = (nl == 0) ? a_s : a_d;
    int row0 = base + hi * 8;
#pragma unroll
    for (int r = 0; r < 8; ++r)
      outp[row0 + r] = c[r] * nrm[row0 + r];
  }
}

// e = leakyrelu(a_s[src] + a_d[dst]); store e; atomic-max into m[dst]
__global__ void k_edge_e(const float* __restrict__ a_s, const float* __restrict__ a_d,
                         const int* __restrict__ src, const int* __restrict__ dst,
                         float* __restrict__ eb, float* __restrict__ m, int n) {
  int i = blockIdx.x * blockDim.x + threadIdx.x;
  if (i >= n) return;
  float e = a_s[src[i]] + a_d[dst[i]];
  e = (e > 0.0f) ? e : 0.2f * e;
  eb[i] = e;
  atomicMaxFloat(&m[dst[i]], e);
}

// ex = exp(e - m[dst]); store ex; s[dst] += ex
__global__ void k_edge_exp(float* __restrict__ eb, const int* __restrict__ dst,
                           const float* __restrict__ m, float* __restrict__ s, int n) {
  int i = blockIdx.x * blockDim.x + threadIdx.x;
  if (i >= n) return;
  int d = dst[i];
  float ex = expf(eb[i] - m[d]);
  eb[i] = ex;
  ATOMIC_ADD_F32(&s[d], ex);
}

// one wave per edge: h[dst] += alpha * x[src] (float4 gather, 4 f32 atomics/lane)
// x and h are L2-resident (25.6 MB each vs 192 MB L2)
__global__ void __launch_bounds__(256)
k_scatter(const float* __restrict__ x, const float* __restrict__ eb,
          const float* __restrict__ s, const int* __restrict__ src,
          const int* __restrict__ dst, float* __restrict__ h, int n_edges) {
  int gtid = blockIdx.x * blockDim.x + threadIdx.x;
  int wave = gtid >> 5;
  int lane = gtid & 31;
  if (wave >= n_edges) return;
  int si = src[wave], di = dst[wave];
  float sv = s[di];
  float alpha = eb[wave] / ((sv > 0.0f) ? sv : 1.0f);
  float4 v = *(const float4*)(x + (size_t)si * DIM + lane * 4);
  float* hp = h + (size_t)di * DIM + lane * 4;
  ATOMIC_ADD_F32(hp + 0, alpha * v.x);
  ATOMIC_ADD_F32(hp + 1, alpha * v.y);
  ATOMIC_ADD_F32(hp + 2, alpha * v.z);
  ATOMIC_ADD_F32(hp + 3, alpha * v.w);
}

// y += h
__global__ void k_accum(const float* __restrict__ h, float* __restrict__ y, int n4) {
  int i = blockIdx.x * blockDim.x + threadIdx.x;
  if (i < n4) {
    float4 a = ((const float4*)h)[i];
    float4 b = ((float4*)y)[i];
    b.x += a.x; b.y += a.y; b.z += a.z; b.w += a.w;
    ((float4*)y)[i] = b;
  }
}

// y *= 1/(order+1)
__global__ void k_scale(float* __restrict__ y, int n4) {
  const float sc = 1.0f / (float)(ORD + 1);
  int i = blockIdx.x * blockDim.x + threadIdx.x;
  if (i < n4) {
    float4 b = ((float4*)y)[i];
    b.x *= sc; b.y *= sc; b.z *= sc; b.w *= sc;
    ((float4*)y)[i] = b;
  }
}

extern "C" void kernel_launch(void* const* d_in, const int* in_sizes, int n_in,
                              void* d_out, int out_size, void* d_ws, size_t ws_size,
                              hipStream_t stream) {
  const float* feats = (const float*)d_in[0];
  const float* w_att = (const float*)d_in[1];
  const int* src = (const int*)d_in[2];
  const int* dst = (const int*)d_in[3];
  float* out = (float*)d_out;

  const size_t ND = (size_t)NODES * DIM;
  // workspace layout (floats): x | h | a_src | a_dst | norm | m | s | e  (~54.6 MB)
  float* ws  = (float*)d_ws;
  float* x   = ws;
  float* h   = x + ND;
  float* a_s = h + ND;
  float* a_d = a_s + NODES;
  float* nrm = a_d + NODES;
  float* m   = nrm + NODES;
  float* s   = m + NODES;
  float* eb  = s + NODES;

  const int B = 256;
  const int nblk_node = (NODES + B - 1) / B;
  const int nblk_edge = (EDGES + B - 1) / B;
  const int nblk_nd4  = (int)((ND / 4 + B - 1) / B);
  const int nblk_dots = (int)(((size_t)((NODES + 15) / 16) * 32 + B - 1) / B);
  const int nblk_scat = (int)(((size_t)EDGES * 32 + B - 1) / B);

  // degree / norm (once), reuse nrm buffer as deg accumulator
  k_zero<<<nblk_node, B, 0, stream>>>(nrm, NODES);
  k_deg<<<nblk_edge, B, 0, stream>>>(dst, nrm, EDGES);
  k_norm<<<nblk_node, B, 0, stream>>>(nrm, NODES);
  k_init<<<nblk_nd4, B, 0, stream>>>(feats, x, out, (int)(ND / 4));

  for (int t = 0; t < ORD; ++t) {
    k_node_dots<<<nblk_dots, B, 0, stream>>>(x, w_att, nrm, a_s, a_d, h, m, s, NODES);
    k_edge_e<<<nblk_edge, B, 0, stream>>>(a_s, a_d, src, dst, eb, m, EDGES);
    k_edge_exp<<<nblk_edge, B, 0, stream>>>(eb, dst, m, s, EDGES);
    k_scatter<<<nblk_scat, B, 0, stream>>>(x, eb, s, src, dst, h, EDGES);
    k_accum<<<nblk_nd4, B, 0, stream>>>(h, out, (int)(ND / 4));
    float* tmp = x; x = h; h = tmp;  // h becomes next x
  }
  k_scale<<<nblk_nd4, B, 0, stream>>>(out, (int)(ND / 4));
}